// HybridRegressionModel_65481071402962
// MI455X (gfx1250) — compile-verified
//
#include <hip/hip_runtime.h>
#include <stdint.h>

#define N_WIRES 10
#define N_OPS   30
#define DIM     1024     // 2^10 amplitudes
#define BSZ     16384
#define APITCH  1025     // LDS row pitch for A tile (bank-conflict free: 1025 % 64 == 1)

// ---------------------------------------------------------------------------
// Host-side bit-exact reproduction of np.random.default_rng(42) op sequence
// (SeedSequence -> PCG64 -> Generator.integers / Generator.choice(Floyd+Lemire))
// Deterministic, recomputed every call; passed to device BY VALUE (capture-safe).
// ---------------------------------------------------------------------------
struct OpsTable { int ty[N_OPS]; int wa[N_OPS]; int wb[N_OPS]; };

namespace npy_rng {

struct Pcg64 {
  unsigned __int128 state, inc;
  int has32; uint32_t buf32;

  void step() {
    const unsigned __int128 mult =
        ((unsigned __int128)2549297995355413924ULL << 64) | 4865540595714422341ULL;
    state = state * mult + inc;
  }
  uint64_t next64() {
    step();
    uint64_t hi = (uint64_t)(state >> 64), lo = (uint64_t)state;
    uint64_t v = hi ^ lo;
    unsigned r = (unsigned)(state >> 122) & 63u;       // top 6 bits
    return (v >> r) | (v << ((64u - r) & 63u));        // rotr64
  }
  uint32_t next32() {                                   // pcg64_next32 buffering
    if (has32) { has32 = 0; return buf32; }
    uint64_t n = next64();
    has32 = 1; buf32 = (uint32_t)(n >> 32);
    return (uint32_t)n;
  }
};

static void seedseq_init(uint32_t entropy, uint32_t pool[4]) {
  const uint32_t INIT_A = 0x43b0d7e5u, MULT_A = 0x931e8875u;
  const uint32_t MIX_L  = 0xca01f9ddu, MIX_R  = 0x4973f715u;
  uint32_t hc = INIT_A;
  auto hashmix = [&](uint32_t v) -> uint32_t {
    v ^= hc; hc *= MULT_A; v *= hc; v ^= v >> 16; return v;
  };
  auto mix = [&](uint32_t x, uint32_t y) -> uint32_t {
    uint32_t r = (x * MIX_L) ^ (y * MIX_R); r ^= r >> 16; return r;
  };
  pool[0] = hashmix(entropy);
  for (int i = 1; i < 4; ++i) pool[i] = hashmix(0u);
  for (int s = 0; s < 4; ++s)
    for (int d = 0; d < 4; ++d)
      if (s != d) pool[d] = mix(pool[d], hashmix(pool[s]));
}

static void seedseq_generate4_u64(const uint32_t pool[4], uint64_t out[4]) {
  const uint32_t INIT_B = 0x8b51f9ddu, MULT_B = 0x58f38dedu;
  uint32_t hc = INIT_B, w[8];
  for (int i = 0; i < 8; ++i) {
    uint32_t v = pool[i & 3];
    v ^= hc; hc *= MULT_B; v *= hc; v ^= v >> 16;
    w[i] = v;
  }
  for (int i = 0; i < 4; ++i)
    out[i] = (uint64_t)w[2 * i] | ((uint64_t)w[2 * i + 1] << 32);  // little-endian pair
}

static void pcg64_from_seed42(Pcg64& g) {
  uint32_t pool[4];  seedseq_init(42u, pool);
  uint64_t sv[4];    seedseq_generate4_u64(pool, sv);
  unsigned __int128 initstate = ((unsigned __int128)sv[0] << 64) | sv[1];
  unsigned __int128 initseq   = ((unsigned __int128)sv[2] << 64) | sv[3];
  g.state = 0; g.inc = (initseq << 1) | 1;
  g.step(); g.state += initstate; g.step();
  g.has32 = 0; g.buf32 = 0;
}

// Generator.integers(): masked rejection on full uint64 draws
static uint64_t masked64(Pcg64& g, uint64_t rng, uint64_t mask) {
  uint64_t v;
  do { v = g.next64() & mask; } while (v > rng);
  return v;
}

// bounded_lemire_uint32: uniform on [0, rng] inclusive
static uint32_t lemire32(Pcg64& g, uint32_t rng) {
  if (rng == 0) return 0;
  const uint32_t rng_excl = rng + 1;
  uint64_t m = (uint64_t)g.next32() * (uint64_t)rng_excl;
  uint32_t leftover = (uint32_t)m;
  if (leftover < rng_excl) {
    const uint32_t threshold = (uint32_t)((0xFFFFFFFFu - rng) % rng_excl);
    while (leftover < threshold) {
      m = (uint64_t)g.next32() * (uint64_t)rng_excl;
      leftover = (uint32_t)m;
    }
  }
  return (uint32_t)(m >> 32);
}

// Generator.choice(10, size=2, replace=False): Floyd's algorithm + shuffle
static void choice10_2(Pcg64& g, int& c, int& t) {
  const uint64_t EMPTY = ~0ull;
  uint64_t hashset[4] = {EMPTY, EMPTY, EMPTY, EMPTY};   // set_size = 4, mask = 3
  int64_t idx[2];
  for (int j = 8; j < 10; ++j) {
    uint64_t val = lemire32(g, (uint32_t)j);            // random_bounded_uint64(.., j, 0, 0)
    uint64_t loc = val & 3;
    while (hashset[loc] != EMPTY && hashset[loc] != val) loc = (loc + 1) & 3;
    if (hashset[loc] == EMPTY) { hashset[loc] = val; idx[j - 8] = (int64_t)val; }
    else {
      loc = (uint64_t)j & 3;
      while (hashset[loc] != EMPTY) loc = (loc + 1) & 3;
      hashset[loc] = (uint64_t)j; idx[j - 8] = j;
    }
  }
  uint64_t jj = lemire32(g, 1);                         // Fisher-Yates, i = 1
  int64_t tmp = idx[1]; idx[1] = idx[jj]; idx[jj] = tmp;
  c = (int)idx[0]; t = (int)idx[1];
}

static void compute_ops(OpsTable& ops) {
  Pcg64 g; pcg64_from_seed42(g);
  for (int k = 0; k < N_OPS; ++k) {
    uint64_t gg = masked64(g, 3, 3);                    // integers(0, 4)
    if (gg == 3) {
      int c, t; choice10_2(g, c, t);
      ops.ty[k] = 3; ops.wa[k] = c; ops.wb[k] = t;
    } else {
      uint64_t w = masked64(g, 9, 15);                  // integers(0, 10)
      ops.ty[k] = (int)gg; ops.wa[k] = (int)w; ops.wb[k] = -1;  // 0=rx,1=ry,2=rz
    }
  }
}

} // namespace npy_rng

// ---------------------------------------------------------------------------
// Kernel 1: build the batch-independent 1024x1024 unitary (30 random ops +
// final RX/RY layer). Block = one basis column simulated in LDS.
//
// Output layout is "pair-major interleaved transposed" for the GEMM B-operand:
// for output row n (= basis index x) and K-group g = k/2:
//   Wp[n*2048 + g*4 + 0] = Re U[x][2g]     Wp[.. + 1] = Re U[x][2g+1]
//   Wp[n*2048 + g*4 + 2] = Im U[x][2g]     Wp[.. + 3] = Im U[x][2g+1]
// so one aligned b128 load per lane per WMMA k-step yields both Re and Im
// fragment pairs in consecutive VGPRs with no deinterleave VALU.
// ---------------------------------------------------------------------------
__device__ __forceinline__ void apply1q(float* re, float* im, int wire,
    float m00r, float m00i, float m01r, float m01i,
    float m10r, float m10i, float m11r, float m11i)
{
  const int p = 9 - wire;          // wire w <-> bit (9-w) of flat index
  const int bit = 1 << p;
  const int lowm = bit - 1;
  for (int t = threadIdx.x; t < DIM / 2; t += blockDim.x) {
    int i0 = ((t & ~lowm) << 1) | (t & lowm);
    int i1 = i0 | bit;
    float a0r = re[i0], a0i = im[i0];
    float a1r = re[i1], a1i = im[i1];
    re[i0] = m00r * a0r - m00i * a0i + m01r * a1r - m01i * a1i;
    im[i0] = m00r * a0i + m00i * a0r + m01r * a1i + m01i * a1r;
    re[i1] = m10r * a0r - m10i * a0i + m11r * a1r - m11i * a1i;
    im[i1] = m10r * a0i + m10i * a0r + m11r * a1i + m11i * a1r;
  }
  __syncthreads();
}

__device__ __forceinline__ void applycnot(float* re, float* im, int cw, int tw) {
  const int pc = 9 - cw, pt = 9 - tw;
  const int lo = pc < pt ? pc : pt;
  const int hi = pc < pt ? pt : pc;
  const int lom = (1 << lo) - 1;
  const int him = (1 << hi) - 1;
  for (int t = threadIdx.x; t < DIM / 4; t += blockDim.x) {
    int x = ((t & ~lom) << 1) | (t & lom);   // insert 0 at lo
    x = ((x & ~him) << 1) | (x & him);       // insert 0 at hi
    int i0 = x | (1 << pc);                  // control = 1, target = 0
    int i1 = i0 | (1 << pt);                 // control = 1, target = 1
    float tr = re[i0]; re[i0] = re[i1]; re[i1] = tr;
    float ti = im[i0]; im[i0] = im[i1]; im[i1] = ti;
  }
  __syncthreads();
}

__global__ __launch_bounds__(128) void qsim_build_unitary(
    const float* __restrict__ rand_params,
    const float* __restrict__ rx_param,
    const float* __restrict__ ry_param,
    float* __restrict__ Wp,
    OpsTable ops)
{
  __shared__ float re[DIM];
  __shared__ float im[DIM];
  const int col = blockIdx.x;
  for (int i = threadIdx.x; i < DIM; i += blockDim.x) {
    re[i] = (i == col) ? 1.0f : 0.0f;
    im[i] = 0.0f;
  }
  __syncthreads();

  for (int k = 0; k < N_OPS; ++k) {
    const int ty = ops.ty[k];
    if (ty == 3) {
      applycnot(re, im, ops.wa[k], ops.wb[k]);
    } else {
      const float h = 0.5f * rand_params[k];
      const float c = cosf(h), s = sinf(h);
      if (ty == 0)       apply1q(re, im, ops.wa[k], c, 0.f, 0.f, -s, 0.f, -s, c, 0.f);   // RX
      else if (ty == 1)  apply1q(re, im, ops.wa[k], c, 0.f, -s, 0.f,  s, 0.f, c, 0.f);   // RY
      else               apply1q(re, im, ops.wa[k], c,  -s, 0.f, 0.f, 0.f, 0.f, c,  s);  // RZ
    }
  }

  const float hx = 0.5f * rx_param[0], hy = 0.5f * ry_param[0];
  const float cx = cosf(hx), sx = sinf(hx);
  const float cy = cosf(hy), sy = sinf(hy);
  for (int w = 0; w < N_WIRES; ++w) {
    apply1q(re, im, w, cx, 0.f, 0.f, -sx, 0.f, -sx, cx, 0.f);   // RX(rx_param)
    apply1q(re, im, w, cy, 0.f, -sy, 0.f,  sy, 0.f, cy, 0.f);   // RY(ry_param)
  }

  // Scattered transpose store (prologue only; 8 MB once).
  // This block owns K=col for every output row n = i.
  const size_t gbase = (size_t)(col >> 1) * 4 + (col & 1);
  for (int i = threadIdx.x; i < DIM; i += blockDim.x) {
    float* p = Wp + (size_t)i * (2 * DIM) + gbase;
    p[0] = re[i];      // Re at group offset (col&1)
    p[2] = im[i];      // Im at group offset 2 + (col&1)
  }
}

// ---------------------------------------------------------------------------
// Kernel 2: fused  A(on-the-fly) x U  ->  probs -> <Z_w> features -> head.
// Block: 16 batch rows, 4 wave32 each owning 16 of the 64 N-tiles.
// Two v_wmma_f32_16x16x4_f32 per k-step fed by one global_load_b128 (B) and
// one ds_load_2addr (A). Epilogue = 16-point Walsh-Hadamard butterfly.
// ---------------------------------------------------------------------------
typedef float v2f __attribute__((ext_vector_type(2)));
typedef float v4f __attribute__((ext_vector_type(4)));
typedef float v8f __attribute__((ext_vector_type(8)));

__global__ __launch_bounds__(128) void qsim_fused_gemm(
    const float* __restrict__ state_batch,
    const float* __restrict__ Wp,
    const float* __restrict__ head_w,
    const float* __restrict__ head_b,
    float* __restrict__ out)
{
  extern __shared__ float smem[];
  float* Alds = smem;                          // [16][APITCH]
  float* feat = smem + 16 * APITCH;            // [16][N_WIRES]
  __shared__ float cs[16][N_WIRES], sn[16][N_WIRES];

  const int tid = threadIdx.x;
  const int b0 = blockIdx.x * 16;

  if (tid < 16 * N_WIRES) {
    const int r = tid / N_WIRES, w = tid % N_WIRES;
    const float h = 0.5f * state_batch[(size_t)(b0 + r) * N_WIRES + w];
    cs[r][w] = cosf(h);
    sn[r][w] = sinf(h);
    feat[tid] = 0.0f;
  }
  __syncthreads();

  // A[b][k] = prod_w (bit_(9-w)(k) ? sin : cos)   (real rank-1 initial state)
  for (int i = tid; i < 16 * DIM; i += 128) {
    const int r = i >> 10, k = i & (DIM - 1);
    float p = 1.0f;
#pragma unroll
    for (int w = 0; w < N_WIRES; ++w)
      p *= ((k >> (9 - w)) & 1) ? sn[r][w] : cs[r][w];
    Alds[r * APITCH + k] = p;
  }
  __syncthreads();

  const int lane = tid & 31;
  const int wv   = tid >> 5;       // wave 0..3
  const int l15  = lane & 15;      // A-row (input) / N-column (output)
  const int half = lane >> 4;      // K-pair select (input) / M+8 select (output)
  const int arow = l15 * APITCH;
  const bool sb3 = (l15 & 8) != 0, sb2 = (l15 & 4) != 0,
             sb1 = (l15 & 2) != 0, sb0 = (l15 & 1) != 0;

  for (int nt = wv; nt < 64; nt += 4) {
    v8f accre = {};
    v8f accim = {};
    // Pair-major interleaved row for output column n = nt*16 + l15:
    // b128 at float offset 2*ka gives {Re[ka],Re[ka+1],Im[ka],Im[ka+1]}.
    const float* bptr = Wp + (size_t)(nt * 16 + l15) * (2 * DIM);
#pragma unroll 4
    for (int k0 = 0; k0 < DIM; k0 += 4) {
      const int ka = k0 + 2 * half;
      if ((k0 & 255) == 0)
        __builtin_prefetch(bptr + 2 * ka + 512, 0, 1);   // global_prefetch_b8
      v2f af = { Alds[arow + ka], Alds[arow + ka + 1] };
      const v4f bc = *reinterpret_cast<const v4f*>(bptr + 2 * ka);
      v2f bfr = { bc[0], bc[1] };
      v2f bfi = { bc[2], bc[3] };
      accre = __builtin_amdgcn_wmma_f32_16x16x4_f32(false, af, false, bfr,
                                                    (short)0, accre, false, false);
      accim = __builtin_amdgcn_wmma_f32_16x16x4_f32(false, af, false, bfi,
                                                    (short)0, accim, false, false);
    }

    // Epilogue: prob = re^2 + im^2 at (row = r + 8*half, x = nt*16 + l15).
    // 16-point WHT over the N index: lane j ends with
    //   W[j] = sum_N (-1)^{popcount(j & N)} p[N].
    // Feature sums needed: j=0 (wires 0..5, sign from nt bits) and
    // j=8,4,2,1 (wires 6..9, single-bit signs).
#pragma unroll
    for (int r = 0; r < 8; ++r) {
      const float p = accre[r] * accre[r] + accim[r] * accim[r];
      float v = p;
      {
        float t = __shfl_xor(v, 1, 32); v = sb0 ? (t - v) : (v + t);
        t = __shfl_xor(v, 2, 32);       v = sb1 ? (t - v) : (v + t);
        t = __shfl_xor(v, 4, 32);       v = sb2 ? (t - v) : (v + t);
        t = __shfl_xor(v, 8, 32);       v = sb3 ? (t - v) : (v + t);
      }
      const int row = r + 8 * half;
      float* f = feat + row * N_WIRES;
      if (l15 == 0) {
#pragma unroll
        for (int w = 0; w < 6; ++w)
          atomicAdd(f + w, ((nt >> (5 - w)) & 1) ? -v : v);
      } else if (l15 == 8) atomicAdd(f + 6, v);
      else if (l15 == 4)   atomicAdd(f + 7, v);
      else if (l15 == 2)   atomicAdd(f + 8, v);
      else if (l15 == 1)   atomicAdd(f + 9, v);
    }
  }
  __syncthreads();

  if (tid < 16) {
    float acc = head_b[0];
#pragma unroll
    for (int w = 0; w < N_WIRES; ++w)
      acc += feat[tid * N_WIRES + w] * head_w[w];
    out[b0 + tid] = acc;
  }
}

// ---------------------------------------------------------------------------
extern "C" void kernel_launch(void* const* d_in, const int* in_sizes, int n_in,
                              void* d_out, int out_size, void* d_ws, size_t ws_size,
                              hipStream_t stream)
{
  (void)in_sizes; (void)n_in; (void)out_size; (void)ws_size;

  const float* state_batch = (const float*)d_in[0];   // [16384][10]
  const float* rand_params = (const float*)d_in[1];   // [30]
  const float* rx_param    = (const float*)d_in[2];   // [1]
  const float* ry_param    = (const float*)d_in[3];   // [1]
  const float* head_w      = (const float*)d_in[4];   // [1][10]
  const float* head_b      = (const float*)d_in[5];   // [1]
  float*       out         = (float*)d_out;           // [16384]

  float* Wp = (float*)d_ws;                           // 8 MB interleaved U

  OpsTable ops;
  npy_rng::compute_ops(ops);

  qsim_build_unitary<<<DIM, 128, 0, stream>>>(rand_params, rx_param, ry_param,
                                              Wp, ops);

  const size_t smem_bytes = (size_t)(16 * APITCH + 16 * N_WIRES) * sizeof(float);
  (void)hipFuncSetAttribute((const void*)qsim_fused_gemm,
                            hipFuncAttributeMaxDynamicSharedMemorySize,
                            (int)smem_bytes);
  qsim_fused_gemm<<<BSZ / 16, 128, smem_bytes, stream>>>(state_batch, Wp,
                                                         head_w, head_b, out);
}